// GAT_70686571758071
// MI455X (gfx1250) — compile-verified
//
#include <hip/hip_runtime.h>
#include <math.h>

// ---------------------------------------------------------------------------
// GAT 2-layer forward for MI455X (gfx1250, wave32).
//  - Layer-1 feature transform on the matrix pipe via V_WMMA_F32_16X16X4_F32.
//  - GEMM staging uses GLOBAL_LOAD_ASYNC_TO_LDS_B128 (ASYNCcnt) when the
//    toolchain exposes the builtin; synchronous fallback otherwise.
//  - x-tile LDS pitch padded 128->132 floats to kill the 16-way bank conflict
//    on A-fragment ds loads (stride 128 floats aliases all 64 banks).
//  - Edge softmax+aggregation in 2 edge passes (atomic max via uint-flip,
//    then unnormalized numerator+denominator); divide at node level.
//  - All accumulators (~60 MB) are L2-resident (192 MB L2).
// ---------------------------------------------------------------------------

typedef __attribute__((ext_vector_type(2))) float v2f;
typedef __attribute__((ext_vector_type(4))) float v4f;
typedef __attribute__((ext_vector_type(8))) float v8f;
typedef __attribute__((ext_vector_type(4))) int   v4i;

#define NEG_SLOPE 0.2f
#define FLIP_NEG_INF 0x007FFFFFu   // flipf(-inf)
#define XPITCH 132                 // 16B-aligned rows, bank-conflict-free frags

#if defined(__has_builtin)
#  if __has_builtin(__builtin_amdgcn_global_load_async_to_lds_b128) && \
      __has_builtin(__builtin_amdgcn_s_wait_asynccnt)
#    define USE_ASYNC_LDS 1
#  endif
#endif

#ifdef USE_ASYNC_LDS
// Signature (from hipcc diagnostic): (v4i AS1*, v4i AS3*, imm offset, imm cpol).
__device__ __forceinline__ void async_b128(void* lds_dst, const void* gsrc) {
    __builtin_amdgcn_global_load_async_to_lds_b128(
        (__attribute__((address_space(1))) v4i*)gsrc,
        (__attribute__((address_space(3))) v4i*)lds_dst,
        /*offset=*/0, /*cpol=*/0);
}
#endif

__device__ __forceinline__ unsigned flipf(float f) {
    unsigned u = __float_as_uint(f);
    return (u & 0x80000000u) ? ~u : (u | 0x80000000u);
}
__device__ __forceinline__ float unflipf(unsigned v) {
    return __uint_as_float((v & 0x80000000u) ? (v ^ 0x80000000u) : ~v);
}

// ---------------------------------------------------------------------------
// Init all accumulator regions (harness poisons ws with 0xAA).
// ---------------------------------------------------------------------------
__global__ void init_kernel(unsigned* __restrict__ emax1, float* __restrict__ denom1,
                            float* __restrict__ out1,
                            unsigned* __restrict__ emax2, float* __restrict__ denom2,
                            float* __restrict__ out2, int N) {
    int i = blockIdx.x * blockDim.x + threadIdx.x;
    if (i < N * 8)  { emax1[i] = FLIP_NEG_INF; denom1[i] = 0.0f; out2[i] = 0.0f; }
    if (i < N * 64) { out1[i] = 0.0f; }
    if (i < N)      { emax2[i] = FLIP_NEG_INF; denom2[i] = 0.0f; }
}

// ---------------------------------------------------------------------------
// Layer-1 GEMM: h1[N,64] = x[N,128] @ W1[128,64] via V_WMMA_F32_16X16X4_F32.
// Block = 128 threads = 4 waves; each wave computes a 16x64 tile.
// LDS: W1 (32KB) + 4 x-tiles of 16xXPITCH (~33KB).
// A frag (16x4 f32): lane L holds M=L%16, K = 2*(L/16)+{0,1} in vgpr {0,1}.
// B frag (4x16 f32): lane L holds N=L%16, K = 2*(L/16)+{0,1} (mirrors A).
// D (16x16 f32, 8 VGPRs): vgpr v -> M = v + 8*(L/16), N = L%16.
// ---------------------------------------------------------------------------
__global__ __launch_bounds__(128)
void gemm1_wmma(const float* __restrict__ x, const float* __restrict__ W1,
                float* __restrict__ h1, int N) {
    __shared__ float lW[128 * 64];          // 32 KB
    __shared__ float lX[4][16 * XPITCH];    // ~33 KB (padded pitch)

    const int tid  = threadIdx.x;
    const int wave = tid >> 5;
    const int lane = tid & 31;
    const int row0 = blockIdx.x * 64 + wave * 16;

    // Stage W1: 8192 floats = 2048 x b128.
#ifdef USE_ASYNC_LDS
    for (int i = tid; i < 2048; i += 128)
        async_b128(&lW[i * 4], &W1[i * 4]);
#else
    for (int i = tid; i < 2048; i += 128)
        ((v4f*)lW)[i] = ((const v4f*)W1)[i];
#endif

    // Stage this wave's 16x128 x-tile (b128 per lane, padded LDS pitch).
    for (int i = lane; i < 512; i += 32) {
        int r  = i >> 5;       // 32 float4 per row
        int c4 = i & 31;
        int gr = row0 + r; if (gr >= N) gr = N - 1;   // clamp; store is guarded
#ifdef USE_ASYNC_LDS
        async_b128(&lX[wave][r * XPITCH + c4 * 4], &x[(size_t)gr * 128 + c4 * 4]);
#else
        *(v4f*)&lX[wave][r * XPITCH + c4 * 4] =
            *(const v4f*)&x[(size_t)gr * 128 + c4 * 4];
#endif
    }
#ifdef USE_ASYNC_LDS
    __builtin_amdgcn_s_wait_asynccnt(0);
#endif
    __syncthreads();

    const int m     = lane & 15;
    const int khalf = (lane >> 4) << 1;   // 0 or 2

    v8f acc0 = {}, acc1 = {}, acc2 = {}, acc3 = {};
    const float* xrow = &lX[wave][m * XPITCH];

    for (int kk = 0; kk < 128; kk += 4) {
        const int kb = kk + khalf;
        v2f a; a.x = xrow[kb]; a.y = xrow[kb + 1];
        const float* w0 = &lW[kb * 64 + m];
        const float* w1 = &lW[(kb + 1) * 64 + m];
        v2f b0; b0.x = w0[0];  b0.y = w1[0];
        v2f b1; b1.x = w0[16]; b1.y = w1[16];
        v2f b2; b2.x = w0[32]; b2.y = w1[32];
        v2f b3; b3.x = w0[48]; b3.y = w1[48];
        acc0 = __builtin_amdgcn_wmma_f32_16x16x4_f32(false, a, false, b0, (short)0, acc0, false, false);
        acc1 = __builtin_amdgcn_wmma_f32_16x16x4_f32(false, a, false, b1, (short)0, acc1, false, false);
        acc2 = __builtin_amdgcn_wmma_f32_16x16x4_f32(false, a, false, b2, (short)0, acc2, false, false);
        acc3 = __builtin_amdgcn_wmma_f32_16x16x4_f32(false, a, false, b3, (short)0, acc3, false, false);
    }

    const int rbase = row0 + ((lane >> 4) << 3);
    #pragma unroll
    for (int v = 0; v < 8; ++v) {
        int gr = rbase + v;
        if (gr < N) {
            float* o = &h1[(size_t)gr * 64 + m];
            o[0]  = acc0[v];
            o[16] = acc1[v];
            o[32] = acc2[v];
            o[48] = acc3[v];
        }
    }
}

// ---------------------------------------------------------------------------
// Per-node attention logits for layer 1: a_src/a_dst [N,8].
// ---------------------------------------------------------------------------
__global__ void att1_kernel(const float* __restrict__ h1,
                            const float* __restrict__ att_src, const float* __restrict__ att_dst,
                            float* __restrict__ as, float* __restrict__ ad, int N) {
    int n = blockIdx.x * blockDim.x + threadIdx.x;
    if (n >= N) return;
    const float* hr = &h1[(size_t)n * 64];
    #pragma unroll
    for (int h = 0; h < 8; ++h) {
        float s = 0.0f, d0 = 0.0f;
        #pragma unroll
        for (int dd = 0; dd < 8; ++dd) {
            float v = hr[h * 8 + dd];
            s  += v * att_src[h * 8 + dd];
            d0 += v * att_dst[h * 8 + dd];
        }
        as[(size_t)n * 8 + h] = s;
        ad[(size_t)n * 8 + h] = d0;
    }
}

// ---------------------------------------------------------------------------
// Edge pass 1, layer 1: per-(dst,head) running max of leaky_relu scores.
// Edges [0,E) from edge_index, edges [E,E+N) are the self-loops.
// ---------------------------------------------------------------------------
__global__ void edge_max1(const long long* __restrict__ src, const long long* __restrict__ dst,
                          int E, int N,
                          const float* __restrict__ as, const float* __restrict__ ad,
                          unsigned* __restrict__ emax) {
    int e = blockIdx.x * blockDim.x + threadIdx.x;
    if (e >= E + N) return;
    // Stream-ahead hint for the sequentially-read index arrays.
    __builtin_prefetch(&src[e + 4096], 0, 0);
    __builtin_prefetch(&dst[e + 4096], 0, 0);
    long long s, d;
    if (e < E) { s = src[e]; d = dst[e]; } else { s = d = (long long)(e - E); }
    #pragma unroll
    for (int h = 0; h < 8; ++h) {
        float v = as[s * 8 + h] + ad[d * 8 + h];
        v = v > 0.0f ? v : NEG_SLOPE * v;
        atomicMax(&emax[d * 8 + h], flipf(v));
    }
}

// ---------------------------------------------------------------------------
// Edge pass 2, layer 1: accumulate unnormalized numerator + denominator.
// ---------------------------------------------------------------------------
__global__ void edge_acc1(const long long* __restrict__ src, const long long* __restrict__ dst,
                          int E, int N,
                          const float* __restrict__ as, const float* __restrict__ ad,
                          const unsigned* __restrict__ emax,
                          const float* __restrict__ h1,
                          float* __restrict__ denom, float* __restrict__ out) {
    int e = blockIdx.x * blockDim.x + threadIdx.x;
    if (e >= E + N) return;
    __builtin_prefetch(&src[e + 4096], 0, 0);
    __builtin_prefetch(&dst[e + 4096], 0, 0);
    long long s, d;
    if (e < E) { s = src[e]; d = dst[e]; } else { s = d = (long long)(e - E); }
    const float* hs = &h1[(size_t)s * 64];
    #pragma unroll
    for (int h = 0; h < 8; ++h) {
        float v = as[s * 8 + h] + ad[d * 8 + h];
        v = v > 0.0f ? v : NEG_SLOPE * v;
        float ex = __expf(v - unflipf(emax[d * 8 + h]));
        atomicAdd(&denom[d * 8 + h], ex);
        float* o = &out[(size_t)d * 64 + h * 8];
        #pragma unroll
        for (int dd = 0; dd < 8; ++dd)
            atomicAdd(&o[dd], ex * hs[h * 8 + dd]);
    }
}

// ---------------------------------------------------------------------------
// Node finalize, layer 1: emb = out/denom + b1 -> d_out; elu(emb) -> elubuf.
// One thread per output element (N*64).
// ---------------------------------------------------------------------------
__global__ void finalize1(const float* __restrict__ out1, const float* __restrict__ denom,
                          const float* __restrict__ b1,
                          float* __restrict__ emb_out, float* __restrict__ elubuf, int N) {
    int i = blockIdx.x * blockDim.x + threadIdx.x;
    if (i >= N * 64) return;
    int n  = i >> 6;
    int hd = i & 63;
    int h  = hd >> 3;
    float v = out1[i] / (denom[(size_t)n * 8 + h] + 1e-16f) + b1[hd];
    emb_out[i] = v;
    elubuf[i]  = v > 0.0f ? v : (__expf(v) - 1.0f);
}

// ---------------------------------------------------------------------------
// Layer-2 GEMM (64x7, too skinny for WMMA) + attention logits (H=1).
// h2 stored with stride 8 for alignment.
// ---------------------------------------------------------------------------
__global__ __launch_bounds__(256)
void gemm2_att2(const float* __restrict__ elubuf, const float* __restrict__ W2,
                const float* __restrict__ att_s2, const float* __restrict__ att_d2,
                float* __restrict__ h2, float* __restrict__ a2s, float* __restrict__ a2d,
                int N) {
    __shared__ float lW[64 * 7];
    for (int i = threadIdx.x; i < 448; i += 256) lW[i] = W2[i];
    __syncthreads();
    int n = blockIdx.x * blockDim.x + threadIdx.x;
    if (n >= N) return;
    float acc[7] = {0.f, 0.f, 0.f, 0.f, 0.f, 0.f, 0.f};
    const float* er = &elubuf[(size_t)n * 64];
    for (int k = 0; k < 64; ++k) {
        float v = er[k];
        #pragma unroll
        for (int c = 0; c < 7; ++c) acc[c] += v * lW[k * 7 + c];
    }
    float s = 0.0f, d0 = 0.0f;
    #pragma unroll
    for (int c = 0; c < 7; ++c) {
        h2[(size_t)n * 8 + c] = acc[c];
        s  += acc[c] * att_s2[c];
        d0 += acc[c] * att_d2[c];
    }
    a2s[n] = s;
    a2d[n] = d0;
}

// ---------------------------------------------------------------------------
// Edge passes, layer 2 (H=1, D=7).
// ---------------------------------------------------------------------------
__global__ void edge_max2(const long long* __restrict__ src, const long long* __restrict__ dst,
                          int E, int N,
                          const float* __restrict__ as, const float* __restrict__ ad,
                          unsigned* __restrict__ emax) {
    int e = blockIdx.x * blockDim.x + threadIdx.x;
    if (e >= E + N) return;
    __builtin_prefetch(&src[e + 4096], 0, 0);
    __builtin_prefetch(&dst[e + 4096], 0, 0);
    long long s, d;
    if (e < E) { s = src[e]; d = dst[e]; } else { s = d = (long long)(e - E); }
    float v = as[s] + ad[d];
    v = v > 0.0f ? v : NEG_SLOPE * v;
    atomicMax(&emax[d], flipf(v));
}

__global__ void edge_acc2(const long long* __restrict__ src, const long long* __restrict__ dst,
                          int E, int N,
                          const float* __restrict__ as, const float* __restrict__ ad,
                          const unsigned* __restrict__ emax,
                          const float* __restrict__ h2,
                          float* __restrict__ denom, float* __restrict__ out) {
    int e = blockIdx.x * blockDim.x + threadIdx.x;
    if (e >= E + N) return;
    __builtin_prefetch(&src[e + 4096], 0, 0);
    __builtin_prefetch(&dst[e + 4096], 0, 0);
    long long s, d;
    if (e < E) { s = src[e]; d = dst[e]; } else { s = d = (long long)(e - E); }
    float v = as[s] + ad[d];
    v = v > 0.0f ? v : NEG_SLOPE * v;
    float ex = __expf(v - unflipf(emax[d]));
    atomicAdd(&denom[d], ex);
    const float* hs = &h2[(size_t)s * 8];
    float* o = &out[(size_t)d * 8];
    #pragma unroll
    for (int c = 0; c < 7; ++c)
        atomicAdd(&o[c], ex * hs[c]);
}

// ---------------------------------------------------------------------------
// Node finalize, layer 2: logits = out/denom + b2, then log_softmax -> d_out.
// ---------------------------------------------------------------------------
__global__ void finalize2(const float* __restrict__ out2, const float* __restrict__ denom2,
                          const float* __restrict__ b2, float* __restrict__ logout, int N) {
    int n = blockIdx.x * blockDim.x + threadIdx.x;
    if (n >= N) return;
    float dn = denom2[n] + 1e-16f;
    float l[7];
    float m = -3.402823466e38f;
    #pragma unroll
    for (int c = 0; c < 7; ++c) {
        l[c] = out2[(size_t)n * 8 + c] / dn + b2[c];
        m = fmaxf(m, l[c]);
    }
    float sum = 0.0f;
    #pragma unroll
    for (int c = 0; c < 7; ++c) sum += __expf(l[c] - m);
    float lse = m + __logf(sum);
    #pragma unroll
    for (int c = 0; c < 7; ++c) logout[(size_t)n * 7 + c] = l[c] - lse;
}

// ---------------------------------------------------------------------------
// Launch
// ---------------------------------------------------------------------------
extern "C" void kernel_launch(void* const* d_in, const int* in_sizes, int n_in,
                              void* d_out, int out_size, void* d_ws, size_t ws_size,
                              hipStream_t stream) {
    const float*     x        = (const float*)d_in[0];
    const long long* eidx     = (const long long*)d_in[1];
    const float*     W1       = (const float*)d_in[2];
    const float*     att_src1 = (const float*)d_in[3];
    const float*     att_dst1 = (const float*)d_in[4];
    const float*     b1       = (const float*)d_in[5];
    const float*     W2       = (const float*)d_in[6];
    const float*     att_src2 = (const float*)d_in[7];
    const float*     att_dst2 = (const float*)d_in[8];
    const float*     b2       = (const float*)d_in[9];

    const int N = in_sizes[0] / 128;
    const int E = in_sizes[1] / 2;
    const long long* src = eidx;
    const long long* dst = eidx + E;

    // Workspace layout (floats).
    float* ws = (float*)d_ws;
    size_t off = 0;
    float*    h1     = ws + off; off += (size_t)N * 64;  // also reused as elu(emb)
    float*    as1    = ws + off; off += (size_t)N * 8;
    float*    ad1    = ws + off; off += (size_t)N * 8;
    unsigned* emax1  = (unsigned*)(ws + off); off += (size_t)N * 8;
    float*    denom1 = ws + off; off += (size_t)N * 8;
    float*    out1   = ws + off; off += (size_t)N * 64;
    float*    h2     = ws + off; off += (size_t)N * 8;
    float*    a2s    = ws + off; off += (size_t)N;
    float*    a2d    = ws + off; off += (size_t)N;
    unsigned* emax2  = (unsigned*)(ws + off); off += (size_t)N;
    float*    denom2 = ws + off; off += (size_t)N;
    float*    out2   = ws + off; off += (size_t)N * 8;

    float* emb_out = (float*)d_out;                  // [N,64]
    float* log_out = (float*)d_out + (size_t)N * 64; // [N,7]

    const int TB = 256;
    const int gN   = (N + TB - 1) / TB;
    const int gN64 = (N * 64 + TB - 1) / TB;
    const int gE   = ((E + N) + TB - 1) / TB;

    init_kernel<<<gN64, TB, 0, stream>>>(emax1, denom1, out1, emax2, denom2, out2, N);

    gemm1_wmma<<<(N + 63) / 64, 128, 0, stream>>>(x, W1, h1, N);
    att1_kernel<<<gN, TB, 0, stream>>>(h1, att_src1, att_dst1, as1, ad1, N);

    edge_max1<<<gE, TB, 0, stream>>>(src, dst, E, N, as1, ad1, emax1);
    edge_acc1<<<gE, TB, 0, stream>>>(src, dst, E, N, as1, ad1, emax1, h1, denom1, out1);
    finalize1<<<gN64, TB, 0, stream>>>(out1, denom1, b1, emb_out, /*elubuf=*/h1, N);

    gemm2_att2<<<gN, TB, 0, stream>>>(h1, W2, att_src2, att_dst2, h2, a2s, a2d, N);
    edge_max2<<<gE, TB, 0, stream>>>(src, dst, E, N, a2s, a2d, emax2);
    edge_acc2<<<gE, TB, 0, stream>>>(src, dst, E, N, a2s, a2d, emax2, h2, denom2, out2);
    finalize2<<<gN, TB, 0, stream>>>(out2, denom2, b2, log_out, N);
}